// SADELoss_81063212745184
// MI455X (gfx1250) — compile-verified
//
#include <hip/hip_runtime.h>
#include <stdint.h>

#define NCLS   50257
#define BATCH  4096
#define TPB    256
#define QPC    1024            // quads per chunk = TPB * 4
#define CHUNKF (QPC * 4)       // floats per chunk = 4096 (16 KB)

#define ALPHA 1.0f
#define BETA  0.5f
#define GAMMA 0.1f

// ---- CDNA5 async Global->LDS (tracked by ASYNCcnt), non-temporal -----------
__device__ __forceinline__ void async_load_b128(const float* sbase, unsigned byte_off,
                                                float* lds_dst) {
    // GVS mode: mem = SGPR base + VGPR 32-bit byte offset. VDST = LDS byte addr.
    unsigned lds_addr = (unsigned)(uintptr_t)(void*)lds_dst; // low 32b of generic = LDS offset
    asm volatile("global_load_async_to_lds_b128 %0, %1, %2 th:TH_LOAD_NT"
                 :: "v"(lds_addr), "v"(byte_off), "s"(sbase)
                 : "memory");
}
#define ASYNC_WAIT(n) asm volatile("s_wait_asynccnt %0" :: "n"(n) : "memory")

// ---- scalar online update (prologue/epilogue only) --------------------------
__device__ __forceinline__ void upd1(float x, int col, float& m, float& s, int& bi) {
    if (x > m) { s = s * __expf(m - x) + 1.0f; m = x; bi = col; }
    else       { s += __expf(x - m); }
}

// ---- K0: zero histogram ----------------------------------------------------
__global__ void k_zero(unsigned* __restrict__ hist) {
    int i = blockIdx.x * TPB + threadIdx.x;
    if (i < NCLS) hist[i] = 0u;
}

// ---- K1: integer histogram of targets (deterministic) ----------------------
__global__ void k_hist(const int* __restrict__ tgt, unsigned* __restrict__ hist) {
    int i = blockIdx.x * TPB + threadIdx.x;
    if (i < BATCH) atomicAdd(&hist[tgt[i]], 1u);
}

// ---- K2: raw class weights + per-block partial sums ------------------------
__global__ __launch_bounds__(TPB)
void k_wraw(const unsigned* __restrict__ hist, const float* __restrict__ class_w,
            const float* __restrict__ class_cnt, const float* __restrict__ total_samples,
            float* __restrict__ w_raw, float* __restrict__ partials) {
    __shared__ float red[TPB];
    int i = blockIdx.x * TPB + threadIdx.x;
    float w = 0.0f;
    if (i < NCLS) {
        float cnt   = class_cnt[i] + (float)hist[i];
        float total = total_samples[0] + (float)BATCH;
        float freq  = cnt / total;
        w = (cnt > 0.0f) ? 1.0f / (freq + 1e-8f) : class_w[i];
        w_raw[i] = w;
    }
    red[threadIdx.x] = w;
    __syncthreads();
    for (int off = TPB / 2; off >= 1; off >>= 1) {
        if ((int)threadIdx.x < off) red[threadIdx.x] += red[threadIdx.x + off];
        __syncthreads();
    }
    if (threadIdx.x == 0) partials[blockIdx.x] = red[0];
}

// ---- K3: fixed-order total of partials ------------------------------------
__global__ void k_wsum(const float* __restrict__ partials, int n, float* __restrict__ wsum) {
    if (threadIdx.x == 0 && blockIdx.x == 0) {
        float s = 0.0f;
        for (int i = 0; i < n; ++i) s += partials[i];
        *wsum = s;
    }
}

// ---- K4: main per-row pass (one block per row) -----------------------------
__global__ __launch_bounds__(TPB)
void k_main(const float* __restrict__ pred, const int* __restrict__ tgt,
            const float* __restrict__ w_raw, const float* __restrict__ wsum_p,
            float* __restrict__ row_out) {
    __shared__ float sbuf[2][CHUNKF];       // 32 KB double buffer
    __shared__ float r_m[TPB], r_s[TPB];
    __shared__ int   r_bi[TPB];

    const int  row     = blockIdx.x;
    const int  t       = threadIdx.x;
    const long rowbase = (long)row * NCLS;                 // float index of row start
    const int  s0      = (int)((4 - (rowbase & 3)) & 3);   // elems until 16B alignment
    const int  nq      = (NCLS - s0) >> 2;                 // full aligned quads in body
    const int  nchunks = (nq + QPC - 1) / QPC;
    const float* base  = pred + rowbase + s0;              // 16B-aligned body start

    // issue chunk c: every wave issues exactly 4 async b128 ops (tail clamped)
    auto issue = [&](int c) {
        float* buf = sbuf[c & 1];
        const int qb = c * QPC;
#pragma unroll
        for (int k = 0; k < 4; ++k) {
            int q  = qb + t + k * TPB;
            int qc = q < nq ? q : (nq - 1);                // clamp: duplicate load, ignored later
            async_load_b128(base, (unsigned)(qc * 16), &buf[(t + k * TPB) * 4]);
        }
    };

    float m = -INFINITY, s = 0.0f;
    int   bi = 0;

    issue(0);
    for (int c = 0; c < nchunks; ++c) {
        if (c + 1 < nchunks) { issue(c + 1); ASYNC_WAIT(4); }  // chunk c done, c+1 in flight
        else                 { ASYNC_WAIT(0); }
        const float4* buf = (const float4*)sbuf[c & 1];
#pragma unroll
        for (int k = 0; k < 4; ++k) {
            int q = c * QPC + t + k * TPB;
            if (q < nq) {
                float4 x = buf[t + k * TPB];
                // quad-level online softmax: exactly 1 exp per element on hot path
                float qm = fmaxf(fmaxf(x.x, x.y), fmaxf(x.z, x.w));
                if (__builtin_expect(qm > m, 0)) {             // rare rescale + argmax update
                    int col = s0 + q * 4;
                    s *= __expf(m - qm);
                    m  = qm;
                    bi = (x.x == qm) ? col
                       : (x.y == qm) ? col + 1
                       : (x.z == qm) ? col + 2 : col + 3;
                }
                s += __expf(x.x - m) + __expf(x.y - m)
                   + __expf(x.z - m) + __expf(x.w - m);
            }
        }
    }

    // thread 0 folds the <=6 unaligned prologue/epilogue elements
    if (t == 0) {
        for (int j = 0; j < s0; ++j)             upd1(pred[rowbase + j], j, m, s, bi);
        for (int j = s0 + nq * 4; j < NCLS; ++j) upd1(pred[rowbase + j], j, m, s, bi);
    }

    r_m[t] = m; r_s[t] = s; r_bi[t] = bi;
    __syncthreads();
    for (int off = TPB / 2; off >= 1; off >>= 1) {
        if (t < off) {
            float m1 = r_m[t],       s1 = r_s[t];
            float m2 = r_m[t + off], s2 = r_s[t + off];
            int   i2 = r_bi[t + off];
            float M  = fmaxf(m1, m2);
            r_s[t] = s1 * __expf(m1 - M) + s2 * __expf(m2 - M);
            if (m2 > m1 || (m2 == m1 && i2 < r_bi[t])) r_bi[t] = i2;
            r_m[t] = M;
        }
        __syncthreads();
    }

    if (t == 0) {
        int   tc   = tgt[row];
        float xt   = pred[rowbase + tc];
        float M    = r_m[0];
        float Z    = r_s[0];
        float logZ = __logf(Z);
        float ce   = (M + logZ) - xt;           // -log_softmax at target
        float maxp = 1.0f / Z;                  // exp(M - M)/Z
        float corr = (r_bi[0] == tc) ? 1.0f : 0.0f;
        float sade = 1.0f + GAMMA * ((1.0f - maxp) + (1.0f - corr));
        float cw   = w_raw[tc] * ((float)NCLS / wsum_p[0]);
        row_out[row] = ce * (ALPHA + BETA * cw + sade);
    }
}

// ---- K5: deterministic mean over rows -> d_out[0] --------------------------
__global__ __launch_bounds__(TPB)
void k_final(const float* __restrict__ rowvals, float* __restrict__ out) {
    __shared__ float red[TPB];
    float s = 0.0f;
    for (int i = threadIdx.x; i < BATCH; i += TPB) s += rowvals[i];
    red[threadIdx.x] = s;
    __syncthreads();
    for (int off = TPB / 2; off >= 1; off >>= 1) {
        if ((int)threadIdx.x < off) red[threadIdx.x] += red[threadIdx.x + off];
        __syncthreads();
    }
    if (threadIdx.x == 0) out[0] = red[0] * (1.0f / (float)BATCH);
}

// ---------------------------------------------------------------------------
static inline size_t align_up(size_t x, size_t a) { return (x + a - 1) & ~(a - 1); }

extern "C" void kernel_launch(void* const* d_in, const int* in_sizes, int n_in,
                              void* d_out, int out_size, void* d_ws, size_t ws_size,
                              hipStream_t stream) {
    const float* pred      = (const float*)d_in[0];   // [4096, 50257] f32
    const int*   tgt       = (const int*)d_in[1];     // [4096] i32
    const float* class_w   = (const float*)d_in[2];   // [50257] f32
    const float* class_cnt = (const float*)d_in[3];   // [50257] f32
    const float* total_s   = (const float*)d_in[4];   // [1] f32
    float*       out       = (float*)d_out;

    char*  ws   = (char*)d_ws;
    size_t off  = 0;
    unsigned* hist   = (unsigned*)(ws + off); off = align_up(off + NCLS * 4, 256);
    float*    w_raw  = (float*)(ws + off);    off = align_up(off + NCLS * 4, 256);
    float*    parts  = (float*)(ws + off);    off = align_up(off + 256 * 4, 256);
    float*    wsum   = (float*)(ws + off);    off = align_up(off + 4, 256);
    float*    rowval = (float*)(ws + off);    off = align_up(off + BATCH * 4, 256);
    (void)ws_size; (void)n_in; (void)in_sizes; (void)out_size;

    const int cblocks = (NCLS + TPB - 1) / TPB;   // 197
    k_zero <<<cblocks, TPB, 0, stream>>>(hist);
    k_hist <<<(BATCH + TPB - 1) / TPB, TPB, 0, stream>>>(tgt, hist);
    k_wraw <<<cblocks, TPB, 0, stream>>>(hist, class_w, class_cnt, total_s, w_raw, parts);
    k_wsum <<<1, 32, 0, stream>>>(parts, cblocks, wsum);
    k_main <<<BATCH, TPB, 0, stream>>>(pred, tgt, w_raw, wsum, rowval);
    k_final<<<1, TPB, 0, stream>>>(rowval, out);
}